// CorticalMap_63599875719265
// MI455X (gfx1250) — compile-verified
//
#include <hip/hip_runtime.h>
#include <cmath>

typedef __attribute__((ext_vector_type(2))) float v2f;
typedef __attribute__((ext_vector_type(8))) float v8f;

#define S_DIM   144
#define C_CH    3
#define K_RF    21
#define TARGET  164
#define D_RF    1323      // C*K*K
#define L2_TAPS 121       // LRI*LRI
#define TAPS_P  124       // padded to multiple of 4
#define ITERS   30

// ---------------------------------------------------------------------------
// latw preprocessing: relu, scale to 12.1/sum, mask by lri_cutoff, renorm.
// One wave (32 lanes) per pixel row of 121 weights; pad row to 124 with zeros.
__global__ __launch_bounds__(256) void prep_latw_kernel(
    const float* __restrict__ lw, const float* __restrict__ cut,
    float* __restrict__ out)
{
    int p    = blockIdx.x * 8 + (threadIdx.x >> 5);
    int lane = threadIdx.x & 31;
    if (p >= S_DIM * S_DIM) return;

    float wv[4], cv[4];
    float s_relu = 0.f, s_reluc = 0.f;
    #pragma unroll
    for (int s = 0; s < 4; ++s) {
        int j = lane + 32 * s;
        float w = 0.f, c = 0.f;
        if (j < L2_TAPS) {
            w = fmaxf(lw[(size_t)p * L2_TAPS + j], 0.f);
            c = cut[j];
        }
        wv[s] = w; cv[s] = c;
        s_relu += w; s_reluc += w * c;
    }
    #pragma unroll
    for (int off = 16; off > 0; off >>= 1) {
        s_relu  += __shfl_xor(s_relu,  off, 32);
        s_reluc += __shfl_xor(s_reluc, off, 32);
    }
    float scale = (121.0f * 0.1f) / s_relu;          // L2*FM / sum(relu)
    float f     = scale / (scale * s_reluc + 1e-7f); // fold both normalizations
    #pragma unroll
    for (int s = 0; s < 4; ++s) {
        int j = lane + 32 * s;
        if (j < TAPS_P) {
            float v = (j < L2_TAPS) ? wv[s] * cv[s] * f : 0.f;
            out[(size_t)p * TAPS_P + j] = v;
        }
    }
}

// ---------------------------------------------------------------------------
// Affinity: one wave per pixel. Streams the 110MB rfs tensor once (coalesced
// within each row); per-row normalization folded algebraically.
__global__ __launch_bounds__(256) void aff_kernel(
    const float* __restrict__ x, const float* __restrict__ rfs,
    const float* __restrict__ env, float* __restrict__ aff)
{
    int p    = blockIdx.x * 8 + (threadIdx.x >> 5);
    int lane = threadIdx.x & 31;
    if (p >= S_DIM * S_DIM) return;
    int py = p / S_DIM, px = p - py * S_DIM;

    float s_r = 0.f, s_re = 0.f, s_rxe = 0.f;
    for (int j = lane; j < D_RF; j += 32) {
        float r  = fmaxf(rfs[(size_t)p * D_RF + j], 0.f);
        float e  = env[j];
        int ch   = j / (K_RF * K_RF);
        int rem  = j - ch * (K_RF * K_RF);
        int ky   = rem / K_RF;
        int kx   = rem - ky * K_RF;
        float xv = x[(size_t)ch * TARGET * TARGET + (size_t)(py + ky) * TARGET + (px + kx)];
        s_r   += r;
        s_re  += r * e;
        s_rxe += r * e * xv;
    }
    #pragma unroll
    for (int off = 16; off > 0; off >>= 1) {
        s_r   += __shfl_xor(s_r,   off, 32);
        s_re  += __shfl_xor(s_re,  off, 32);
        s_rxe += __shfl_xor(s_rxe, off, 32);
    }
    if (lane == 0) {
        float scale = (1323.0f * 0.1f) / s_r;   // D*FM / sum(relu)
        aff[p] = (scale * s_rxe) / (scale * s_re + 1e-7f);
    }
}

// ---------------------------------------------------------------------------
__global__ void zero_kernel(float* __restrict__ p, int n)
{
    int i = blockIdx.x * blockDim.x + threadIdx.x;
    if (i < n) p[i] = 0.f;
}

// ---------------------------------------------------------------------------
// Extract element idx (0..7) from a v8f accumulator without scratch spills.
__device__ __forceinline__ float sel8(v8f c, int idx)
{
    float r = c[0];
    r = (idx == 1) ? c[1] : r;
    r = (idx == 2) ? c[2] : r;
    r = (idx == 3) ? c[3] : r;
    r = (idx == 4) ? c[4] : r;
    r = (idx == 5) ? c[5] : r;
    r = (idx == 6) ? c[6] : r;
    r = (idx == 7) ? c[7] : r;
    return r;
}

// ---------------------------------------------------------------------------
// One recurrent step. Block = 256 threads = 8 waves; each block owns an
// 8-row x 16-col output strip. Each wave computes its 16-pixel row with
// V_WMMA_F32_16X16X4_F32:
//   lp: A = lat neighborhood, B = broadcast sre weights  -> all D columns = lp
//   ln: A = dilated lat tiles, B = per-pixel latw (cols) -> diag(D) = ln
// Taps 121..123 are padding: B is zero there (latwp rows and lds_sre are
// zero-padded), and the offset table clamps them to tap 120 so A reads stay
// in range -> branch-free inner loop, EXEC uniformly all-ones for WMMA.
__global__ __launch_bounds__(256) void step_kernel(
    const float* __restrict__ lat_in, float* __restrict__ lat_out,
    const float* __restrict__ latwp, const float* __restrict__ aff,
    const float* __restrict__ sre)
{
    __shared__ __align__(16) float lds_ln[28 * 36]; // zero-pad, reach +-10 (dilated)
    __shared__ __align__(16) float lds_lp[18 * 26]; // reflect-pad, reach +-5 (sre)
    __shared__ __align__(16) float lds_sre[TAPS_P]; // sre * EXC, zero-pad to 124
    __shared__ __align__(16) int   lds_off[TAPS_P]; // packed (lnOff | lpOff<<16)

    const int tid = threadIdx.x;
    const int gx0 = blockIdx.x * 16;
    const int gy0 = blockIdx.y * 8;

    for (int i = tid; i < 28 * 36; i += 256) {
        int yy = i / 36, xx = i - yy * 36;
        int gy = gy0 + yy - 10, gx = gx0 + xx - 10;
        float v = 0.f;
        if (gy >= 0 && gy < S_DIM && gx >= 0 && gx < S_DIM)
            v = lat_in[gy * S_DIM + gx];
        lds_ln[i] = v;
    }
    for (int i = tid; i < 18 * 26; i += 256) {
        int yy = i / 26, xx = i - yy * 26;
        int gy = gy0 + yy - 5, gx = gx0 + xx - 5;
        gy = (gy < 0) ? -gy : ((gy >= S_DIM) ? 2 * S_DIM - 2 - gy : gy);
        gx = (gx < 0) ? -gx : ((gx >= S_DIM) ? 2 * S_DIM - 2 - gx : gx);
        lds_lp[i] = lat_in[gy * S_DIM + gx];
    }
    if (tid < TAPS_P) {
        lds_sre[tid] = (tid < L2_TAPS) ? sre[tid] * 2.0f : 0.f;  // fold EXC
        int t  = (tid < L2_TAPS) ? tid : (L2_TAPS - 1);          // clamp padding
        int ky = t / 11, kx = t - ky * 11;
        lds_off[tid] = (2 * ky * 36 + 2 * kx) | ((ky * 26 + kx) << 16);
    }
    __syncthreads();

    const int lane = tid & 31;
    const int r    = tid >> 5;           // strip row 0..7 handled by this wave
    const int m    = lane & 15;          // matrix row (A) / column (B)
    const int koff = (lane >> 4) << 1;   // K sub-offset: lanes 0-15 -> 0, 16-31 -> 2
    const int gy   = gy0 + r;
    const int p_m  = gy * S_DIM + gx0 + m;

    const int   base_ln = r * 36 + m;    // element offset of pixel (r,m) in lds_ln
    const int   base_lp = r * 26 + m;    // element offset of pixel (r,m) in lds_lp
    const float* __restrict__ wrow = latwp + (size_t)p_m * TAPS_P;

    v8f c_ln = {};
    v8f c_lp = {};
    #pragma unroll 4
    for (int k0 = 0; k0 < TAPS_P; k0 += 4) {
        const int t0 = k0 + koff;        // even -> 8B-aligned b64 loads below

        const int2 po = *reinterpret_cast<const int2*>(&lds_off[t0]);
        v2f a_ln, a_lp, b_ln, b_lp;
        a_ln.x = lds_ln[base_ln + (po.x & 0xffff)];
        a_ln.y = lds_ln[base_ln + (po.y & 0xffff)];
        a_lp.x = lds_lp[base_lp + ((unsigned)po.x >> 16)];
        a_lp.y = lds_lp[base_lp + ((unsigned)po.y >> 16)];

        const float2 bw = *reinterpret_cast<const float2*>(wrow + t0);
        b_ln.x = bw.x; b_ln.y = bw.y;
        const float2 ww = *reinterpret_cast<const float2*>(&lds_sre[t0]);
        b_lp.x = ww.x; b_lp.y = ww.y;

        c_ln = __builtin_amdgcn_wmma_f32_16x16x4_f32(false, a_ln, false, b_ln,
                                                     (short)0, c_ln, false, false);
        c_lp = __builtin_amdgcn_wmma_f32_16x16x4_f32(false, a_lp, false, b_lp,
                                                     (short)0, c_lp, false, false);
    }

    // Diagonal D[m,m] lives in lanes 0-7 (vgpr = lane) and 24-31 (vgpr = lane-24).
    if (lane < 8 || lane >= 24) {
        int idx = lane & 7;
        int mm  = (lane < 8) ? lane : (lane - 16);
        float ln_v = sel8(c_ln, idx);
        float lp_v = sel8(c_lp, idx);   // all columns equal -> same indexing
        int pp = gy * S_DIM + gx0 + mm;
        float v = lp_v - ln_v + aff[pp];
        v = fmaxf(v, 0.f);
        lat_out[pp] = tanhf(v);         // SAT = 1.0
    }
}

// ---------------------------------------------------------------------------
extern "C" void kernel_launch(void* const* d_in, const int* in_sizes, int n_in,
                              void* d_out, int out_size, void* d_ws, size_t ws_size,
                              hipStream_t stream)
{
    (void)in_sizes; (void)n_in; (void)out_size; (void)ws_size;
    const float* x    = (const float*)d_in[0];   // (1,3,164,164)
    const float* rfs  = (const float*)d_in[1];   // (20736,1323,1)
    const float* lw   = (const float*)d_in[2];   // (20736,121,1)
    const float* env  = (const float*)d_in[3];   // (1323,)
    const float* sre  = (const float*)d_in[4];   // (1,1,11,11)
    const float* cut  = (const float*)d_in[5];   // (121,)
    float* out = (float*)d_out;                  // (1,1,144,144)

    const int NPIX = S_DIM * S_DIM;              // 20736
    char* ws = (char*)d_ws;
    size_t off = 0;
    auto alloc = [&](size_t bytes) {
        void* p = ws + off;
        off = (off + bytes + 255) & ~(size_t)255;
        return p;
    };
    float* latwp = (float*)alloc((size_t)NPIX * TAPS_P * sizeof(float)); // ~10.3MB
    float* affb  = (float*)alloc((size_t)NPIX * sizeof(float));
    float* s0    = (float*)alloc((size_t)NPIX * sizeof(float));
    float* s1    = (float*)alloc((size_t)NPIX * sizeof(float));

    prep_latw_kernel<<<NPIX / 8, 256, 0, stream>>>(lw, cut, latwp);
    aff_kernel<<<NPIX / 8, 256, 0, stream>>>(x, rfs, env, affb);
    zero_kernel<<<NPIX / 256, 256, 0, stream>>>(s0, NPIX);

    const float* cur = s0;
    for (int i = 0; i < ITERS; ++i) {
        float* dst = (i == ITERS - 1) ? out : ((i & 1) ? s0 : s1);
        step_kernel<<<dim3(S_DIM / 16, S_DIM / 8), 256, 0, stream>>>(
            cur, dst, latwp, affb, sre);
        cur = dst;
    }
}